// Net_76562087018570
// MI455X (gfx1250) — compile-verified
//
#include <hip/hip_runtime.h>
#include <stdint.h>

#define DROWS 4096
#define NCOLS 8192
#define N4    (NCOLS / 4)
#define N_ITERS 10
#define ROWS_PER_CHUNK 64

typedef __attribute__((ext_vector_type(2))) float v2f;
typedef __attribute__((ext_vector_type(8))) float v8f;
typedef __attribute__((ext_vector_type(4))) unsigned int v4u;
typedef __attribute__((ext_vector_type(8))) int v8i;
typedef __attribute__((ext_vector_type(4))) int v4i;

// Monotonic float <-> uint encoding so unsigned atomicMin/Max implement float min/max.
__device__ __forceinline__ unsigned int encf(float f) {
    unsigned int u = __float_as_uint(f);
    return (u & 0x80000000u) ? ~u : (u | 0x80000000u);
}
__device__ __forceinline__ float decf(unsigned int e) {
    unsigned int u = (e & 0x80000000u) ? (e & 0x7FFFFFFFu) : ~e;
    return __uint_as_float(u);
}

// ---------------- Kernel 0: init workspace (encoded min=+max_enc, max=0) ----------------
__global__ void __launch_bounds__(256) init_minmax(unsigned int* ws) {
    int i = blockIdx.x * 256 + threadIdx.x;
    if (i < NCOLS) {
        ws[i]         = 0xFFFFFFFFu;  // enc(min slot) start at +inf-equivalent
        ws[NCOLS + i] = 0x00000000u;  // enc(max slot) start at -inf-equivalent
    }
}

// ---------------- Kernel 1: per-column min/max over all rows ----------------
__global__ void __launch_bounds__(256) colminmax(const float4* __restrict__ x4,
                                                 unsigned int* __restrict__ ws) {
    int c4 = blockIdx.x * 256 + threadIdx.x;      // float4 column index
    int r0 = blockIdx.y * ROWS_PER_CHUNK;
    float4 v0 = x4[(size_t)r0 * N4 + c4];
    float mnx = v0.x, mny = v0.y, mnz = v0.z, mnw = v0.w;
    float mxx = v0.x, mxy = v0.y, mxz = v0.z, mxw = v0.w;
    for (int r = 1; r < ROWS_PER_CHUNK; ++r) {
        float4 v = x4[(size_t)(r0 + r) * N4 + c4];
        mnx = fminf(mnx, v.x); mxx = fmaxf(mxx, v.x);
        mny = fminf(mny, v.y); mxy = fmaxf(mxy, v.y);
        mnz = fminf(mnz, v.z); mxz = fmaxf(mxz, v.z);
        mnw = fminf(mnw, v.w); mxw = fmaxf(mxw, v.w);
    }
    int c = 4 * c4;
    atomicMin(&ws[c + 0], encf(mnx)); atomicMax(&ws[NCOLS + c + 0], encf(mxx));
    atomicMin(&ws[c + 1], encf(mny)); atomicMax(&ws[NCOLS + c + 1], encf(mxy));
    atomicMin(&ws[c + 2], encf(mnz)); atomicMax(&ws[NCOLS + c + 2], encf(mxz));
    atomicMin(&ws[c + 3], encf(mnw)); atomicMax(&ws[NCOLS + c + 3], encf(mxw));
}

// ---------------- Kernel 2: one block per row; 10 iterations in registers ----------------
__global__ void __launch_bounds__(256) row_iters(const float* __restrict__ x,
                                                 const unsigned int* __restrict__ ws,
                                                 float* __restrict__ out) {
    __shared__ float tile[NCOLS];      // 32 KB staged row
    __shared__ float partials[16];     // 2 half-wave sums per wave, 8 waves

    const int tid = threadIdx.x;
    const int row = blockIdx.x;

    // --- TDM: DMA the 32KB row into LDS (one wave issues; EXEC ignored by TDM) ---
    if (tid < 32) {
        unsigned long long ga = (unsigned long long)(uintptr_t)(x + (size_t)row * NCOLS);
        unsigned int lds = (unsigned int)(uintptr_t)(&tile[0]);
        v4u g0;
        g0.x = 1u;                                             // count=1 (valid descriptor)
        g0.y = lds;                                            // lds_addr
        g0.z = (unsigned int)(ga & 0xFFFFFFFFu);               // global_addr[31:0]
        g0.w = (unsigned int)((ga >> 32) & 0x01FFFFFFu) | (2u << 30);  // addr[56:32] | type=2
        v8i g1;
        g1[0] = (int)(2u << 16);                               // data_size=2 (4 bytes)
        g1[1] = (int)(((unsigned)NCOLS & 0xFFFFu) << 16);      // tensor_dim0[15:0] @63:48
        g1[2] = (int)(1u << 16);                               // dim0 hi=0 | tensor_dim1=1
        g1[3] = (int)((unsigned)NCOLS << 16);                  // dim1 hi=0 | tile_dim0=8192
        g1[4] = 1;                                             // tile_dim1=1, tile_dim2=0
        g1[5] = NCOLS;                                         // tensor_dim0_stride lo32
        g1[6] = 0;
        g1[7] = 0;
        v4i z4 = {0, 0, 0, 0};
        v8i z8 = {0, 0, 0, 0, 0, 0, 0, 0};
        __builtin_amdgcn_tensor_load_to_lds(g0, g1, z4, z4, z8, 0);
        __builtin_amdgcn_s_wait_tensorcnt(0);
    }
    __syncthreads();

    // --- Pull 32 columns/thread into registers; normalize with column min/range ---
    float a[32], mn[32], sr[32];
#pragma unroll
    for (int k = 0; k < 8; ++k) {
        int c4 = tid + (k << 8);                    // float4 column index
        float4 v = ((const float4*)tile)[c4];
        uint4 emn = ((const uint4*)ws)[c4];
        uint4 emx = ((const uint4*)(ws + NCOLS))[c4];
        float l0 = decf(emn.x), l1 = decf(emn.y), l2 = decf(emn.z), l3 = decf(emn.w);
        float h0 = decf(emx.x), h1 = decf(emx.y), h2 = decf(emx.z), h3 = decf(emx.w);
        float r0 = h0 - l0; r0 = (r0 == 0.0f) ? 1.0f : r0;
        float r1 = h1 - l1; r1 = (r1 == 0.0f) ? 1.0f : r1;
        float r2 = h2 - l2; r2 = (r2 == 0.0f) ? 1.0f : r2;
        float r3 = h3 - l3; r3 = (r3 == 0.0f) ? 1.0f : r3;
        int j = 4 * k;
        mn[j + 0] = l0; sr[j + 0] = r0; a[j + 0] = (v.x - l0) / r0;
        mn[j + 1] = l1; sr[j + 1] = r1; a[j + 1] = (v.y - l1) / r1;
        mn[j + 2] = l2; sr[j + 2] = r2; a[j + 2] = (v.z - l2) / r2;
        mn[j + 3] = l3; sr[j + 3] = r3; a[j + 3] = (v.w - l3) / r3;
    }

    const int lane = tid & 31;
    const int wave = tid >> 5;

    // --- 10 iterations entirely in registers; row-sum via WMMA wave reduction ---
    for (int it = 0; it < N_ITERS; ++it) {
        float ls = 0.0f;
#pragma unroll
        for (int j = 0; j < 32; ++j) ls += a[j];

        // Wave reduction: A(16x4) = lane values in K=0 (lanes 0-15) / K=2 (lanes 16-31),
        // K=1,3 zero; B = ones(4x16). D accumulators hold per-row sums:
        // lanes 0-15 get rows 0-7 (val[m]+val[m+16]), lanes 16-31 get rows 8-15.
        v2f A; A.x = ls; A.y = 0.0f;
        v2f B; B.x = 1.0f; B.y = 1.0f;
        v8f C = {0.0f, 0.0f, 0.0f, 0.0f, 0.0f, 0.0f, 0.0f, 0.0f};
        v8f Dm = __builtin_amdgcn_wmma_f32_16x16x4_f32(
            false, A, false, B, (short)0, C, false, false);
        float hsum = 0.0f;
#pragma unroll
        for (int j = 0; j < 8; ++j) hsum += Dm[j];

        if (lane == 0)  partials[2 * wave + 0] = hsum;  // sum of lanes {0..7, 16..23}
        if (lane == 16) partials[2 * wave + 1] = hsum;  // sum of lanes {8..15, 24..31}
        __syncthreads();
        float tot = 0.0f;
#pragma unroll
        for (int w = 0; w < 16; ++w) tot += partials[w];
        __syncthreads();  // protect partials before next iteration's writes

        float mean = tot * (1.0f / (float)NCOLS);
        float add = (it == 0) ? 0.001f : 0.0f;
#pragma unroll
        for (int j = 0; j < 32; ++j) {
            float p = a[j] - mean;                // |p|*sign(p) == p
            a[j] = a[j] * (1.0f - p) + add;
        }
    }

    // --- Denormalize and store ---
#pragma unroll
    for (int k = 0; k < 8; ++k) {
        int c4 = tid + (k << 8);
        int j = 4 * k;
        float4 o;
        o.x = a[j + 0] * sr[j + 0] + mn[j + 0];
        o.y = a[j + 1] * sr[j + 1] + mn[j + 1];
        o.z = a[j + 2] * sr[j + 2] + mn[j + 2];
        o.w = a[j + 3] * sr[j + 3] + mn[j + 3];
        ((float4*)out)[(size_t)row * N4 + c4] = o;
    }
}

extern "C" void kernel_launch(void* const* d_in, const int* in_sizes, int n_in,
                              void* d_out, int out_size, void* d_ws, size_t ws_size,
                              hipStream_t stream) {
    const float* x = (const float*)d_in[0];
    float* out = (float*)d_out;
    unsigned int* ws = (unsigned int*)d_ws;   // [0..N): enc min, [N..2N): enc max  (64 KB)

    init_minmax<<<NCOLS / 256, 256, 0, stream>>>(ws);

    dim3 g1(N4 / 256, DROWS / ROWS_PER_CHUNK);   // (8, 64)
    colminmax<<<g1, 256, 0, stream>>>((const float4*)x, ws);

    row_iters<<<DROWS, 256, 0, stream>>>(x, ws, out);
}